// HungarianLoss_6459630813759
// MI455X (gfx1250) — compile-verified
//
#include <hip/hip_runtime.h>
#include <stdint.h>

#define N 256
#define BATCH 64

typedef unsigned int u32x4 __attribute__((ext_vector_type(4)));
typedef unsigned int u32x8 __attribute__((ext_vector_type(8)));

// Dynamic LDS layout (bytes):
//   cost  : N*N*4            = 262144
//   u,v,minv (double[260]ea) + wval(double[8])  = 6304
//   p,way,used (int[260]ea)  + widx(int[8])     = 3152
static constexpr size_t SMEM_BYTES = (size_t)N * N * 4 + (260 * 3 + 8) * 8 + (260 * 3 + 8) * 4;

__global__ __launch_bounds__(256) void lap_kernel(const float* __restrict__ Dmat,
                                                  double* __restrict__ partial)
{
    extern __shared__ char smem[];
    float*  cost = (float*)smem;
    double* u    = (double*)(smem + (size_t)N * N * 4);
    double* v    = u + 260;
    double* minv = v + 260;
    double* wval = minv + 260;          // 8 per-wave reduction slots
    int*    p    = (int*)(wval + 8);
    int*    way  = p + 260;
    int*    used = way + 260;
    int*    widx = used + 260;          // 8

    const int t    = threadIdx.x;       // 0..255
    const int b    = blockIdx.x;        // batch
    const int j    = t + 1;             // column 1..256 (1-based like reference)
    const int lane = t & 31;
    const int wave = t >> 5;

    // ---- TDM: DMA the whole 256x256 fp32 cost tile into LDS (async) ----
    if (t == 0) {
        uint64_t gaddr    = (uint64_t)(const void*)(Dmat + (size_t)b * N * N);
        unsigned lds_base = __builtin_amdgcn_groupstaticsize(); // dynamic LDS starts here
        u32x4 g0;
        g0[0] = 1u;                                  // count=1, user descriptor
        g0[1] = lds_base;                            // lds_addr (bytes)
        g0[2] = (unsigned)(gaddr & 0xFFFFFFFFu);     // global_addr[31:0]
        g0[3] = (unsigned)((gaddr >> 32) & 0x1FFFFFFu) | (2u << 30); // addr[56:32] | type=2
        u32x8 g1;
        g1[0] = (2u << 16);                          // data_size=2 -> 4 bytes
        g1[1] = ((unsigned)N & 0xFFFFu) << 16;       // tensor_dim0 lo16 (atomic_barrier=0)
        g1[2] = ((unsigned)N & 0xFFFFu) << 16;       // dim0 hi16=0 | tensor_dim1 lo16
        g1[3] = ((unsigned)N & 0xFFFFu) << 16;       // dim1 hi16=0 | tile_dim0=256
        g1[4] = (unsigned)N;                         // tile_dim1=256 | tile_dim2=0
        g1[5] = (unsigned)N;                         // tensor_dim0_stride lo32 = 256 elems
        g1[6] = 0u;                                  // stride hi | dim1_stride lo
        g1[7] = 0u;
        asm volatile("tensor_load_to_lds %0, %1" :: "s"(g0), "s"(g1) : "memory");
    }

    // Init potentials/matching while the TDM transfer is in flight.
    u[j] = 0.0; v[j] = 0.0; p[j] = 0;
    if (t == 0) { u[0] = 0.0; v[0] = 0.0; p[0] = 0; }

    if (t == 0) { __builtin_amdgcn_s_wait_tensorcnt(0); }
    __syncthreads();

    const double INFD = __builtin_huge_val();

    for (int i = 1; i <= N; ++i) {
        // per-row init
        minv[j] = INFD;
        used[j] = 0;
        if (t == 0) { p[0] = i; used[0] = 1; }
        __syncthreads();

        int j0 = 0;
        while (true) {
            const int    i0    = p[j0];
            const double u_i0  = u[i0];
            const int    freej = !used[j];
            double cand = INFD;
            if (freej) {
                double cur = (double)cost[(i0 - 1) * N + (j - 1)] - u_i0 - v[j];
                if (cur < minv[j]) { minv[j] = cur; way[j] = j0; }
                cand = minv[j];
            }
            // wave32 butterfly argmin, lowest-index tie-break (== np.argmin)
            double bv = cand; int bj = j;
            for (int off = 16; off > 0; off >>= 1) {
                double ov = __shfl_xor(bv, off, 32);
                int    oj = __shfl_xor(bj, off, 32);
                if (ov < bv || (ov == bv && oj < bj)) { bv = ov; bj = oj; }
            }
            if (lane == 0) { wval[wave] = bv; widx[wave] = bj; }
            __syncthreads();
            double delta = wval[0]; int j1 = widx[0];
            for (int w = 1; w < 8; ++w) {
                double ov = wval[w]; int oj = widx[w];
                if (ov < delta || (ov == delta && oj < j1)) { delta = ov; j1 = oj; }
            }
            // dual updates (disjoint LDS addresses across threads)
            if (!freej) { v[j] -= delta; u[p[j]] += delta; }
            else        { minv[j] -= delta; if (j == j1) used[j] = 1; }
            if (t == 0) { u[p[0]] += delta; /* v[0] never read */ v[0] -= delta; }
            __syncthreads();
            j0 = j1;
            if (p[j0] == 0) break;  // uniform: all threads read same LDS scalar
        }
        // augment alternating path (short pointer chase, single thread)
        if (t == 0) {
            int jj = j0;
            while (jj) { int jn = way[jj]; p[jj] = p[jn]; jj = jn; }
        }
        __syncthreads();
    }

    // matched sum: sum_j D[b][p[j]-1][j-1]
    double term = (double)cost[(p[j] - 1) * N + (j - 1)];
    for (int off = 16; off > 0; off >>= 1) term += __shfl_xor(term, off, 32);
    if (lane == 0) wval[wave] = term;
    __syncthreads();
    if (t == 0) {
        double s = 0.0;
        for (int w = 0; w < 8; ++w) s += wval[w];
        partial[b] = s;
    }
}

// Deterministic fixed-order final reduction (no float atomics).
__global__ void finalize_kernel(const double* __restrict__ partial, float* __restrict__ out)
{
    double s = 0.0;
    for (int b = 0; b < BATCH; ++b) s += partial[b];
    out[0] = (float)(s / (double)((size_t)N * BATCH));
}

extern "C" void kernel_launch(void* const* d_in, const int* in_sizes, int n_in,
                              void* d_out, int out_size, void* d_ws, size_t ws_size,
                              hipStream_t stream) {
    const float* Dmat    = (const float*)d_in[0];
    double*      partial = (double*)d_ws;          // 64 doubles of scratch
    (void)in_sizes; (void)n_in; (void)out_size; (void)ws_size;

    // Opt in to >64KB dynamic LDS (MI455X: 320KB per WGP).
    hipFuncSetAttribute((const void*)lap_kernel,
                        hipFuncAttributeMaxDynamicSharedMemorySize,
                        (int)SMEM_BYTES);

    lap_kernel<<<BATCH, 256, SMEM_BYTES, stream>>>(Dmat, partial);
    finalize_kernel<<<1, 1, 0, stream>>>(partial, (float*)d_out);
}